// PositionConditionerBlock_33380485825341
// MI455X (gfx1250) — compile-verified
//
#include <hip/hip_runtime.h>
#include <hip/hip_bf16.h>
#include <math.h>

typedef __attribute__((ext_vector_type(2))) float v2f;
typedef __attribute__((ext_vector_type(8))) float v8f;

#define S_LEN 2048
#define NDIM  128
#define H3    24      // 3*H heads total
#define CH    32      // channels per head
#define KQROW 768     // 24*32, row stride of Kbuf
#define VROW  3072    // 24*128, row stride of Vbuf

__device__ __forceinline__ v8f wmma4(v2f a, v2f b, v8f c) {
    // D = A(16x4 f32) * B(4x16 f32) + C(16x16 f32), full fp32 precision
    return __builtin_amdgcn_wmma_f32_16x16x4_f32(false, a, false, b, (short)0, c, false, false);
}

// ---------------------------------------------------------------------------
// GEMM: Out(M=2048 x N) = A(2048 x 128) @ W(N x 128)^T + bias(N)
// One wave per (n-tile, i-group of 8 i-tiles); B (weight) slices held in regs.
// ---------------------------------------------------------------------------
__global__ __launch_bounds__(256) void gemm_wmma_kernel(
        const float* __restrict__ A, const float* __restrict__ W,
        const float* __restrict__ bias, float* __restrict__ Out, int N) {
    const int tid  = threadIdx.x;
    const int lane = tid & 31;
    const int wave = tid >> 5;
    const int hl   = lane >> 4;   // half-wave select
    const int ln   = lane & 15;
    const int ntiles = N >> 4;
    const int g  = blockIdx.x * 8 + wave;
    const int nt = g % ntiles;    // 16-wide output-column tile
    const int ig = g / ntiles;    // group of 8 16-row tiles

    // B operand: slice s VGPR0 = W^T[4s+2hl][n], VGPR1 = W^T[4s+2hl+1][n]
    v2f wb[32];
    const float* Wp = W + (size_t)(nt * 16 + ln) * NDIM + 2 * hl;
#pragma unroll
    for (int s = 0; s < 32; ++s) wb[s] = *(const v2f*)(Wp + s * 4);

    const float bv = bias[nt * 16 + ln];   // bias depends only on column n
    v8f cinit;
#pragma unroll
    for (int r = 0; r < 8; ++r) cinit[r] = bv;

    for (int t2 = 0; t2 < 8; ++t2) {
        const int i0 = (ig * 8 + t2) * 16;
        v8f c = cinit;
        const float* Ap = A + (size_t)(i0 + ln) * NDIM + 2 * hl;
#pragma unroll
        for (int s = 0; s < 32; ++s) {
            v2f a = *(const v2f*)(Ap + s * 4);
            c = wmma4(a, wb[s], c);
        }
        // C/D layout: VGPR r, lanes(hl) -> row r+8*hl, col = ln
        float* Op = Out + (size_t)i0 * N + nt * 16 + ln;
#pragma unroll
        for (int r = 0; r < 8; ++r) Op[(size_t)(r + 8 * hl) * N] = c[r];
    }
}

// ---------------------------------------------------------------------------
// Small projections (aux: K=16, rot: K=4) + scatter nodes_kq into Kbuf / QT.
// Kbuf: (2048, 24, 32) row-major.  QT: (24, 32, 2048) (transposed for B loads)
// ---------------------------------------------------------------------------
__global__ __launch_bounds__(256) void kq_small_kernel(
        const float* __restrict__ aux, const float* __restrict__ rot,
        const float* __restrict__ W_aux, const float* __restrict__ b_aux,
        const float* __restrict__ W_rot, const float* __restrict__ KQn,
        float* __restrict__ Kbuf, float* __restrict__ QT) {
    const int t  = blockIdx.x * 256 + threadIdx.x;   // t < 2048*512
    const int i  = t >> 9;
    const int r  = t & 511;        // row of the (512 x *) projection
    const int hh = r >> 6;         // head within source (0..7)
    const int cc = r & 63;         // 0..31 -> k, 32..63 -> q

    const float nv = KQn[t];       // nodes projection (t == i*512 + r)
    float av = b_aux[r];
#pragma unroll
    for (int x = 0; x < 16; ++x) av += aux[i * 16 + x] * W_aux[r * 16 + x];
    float rv = 0.f;
#pragma unroll
    for (int x = 0; x < 4; ++x) rv += rot[i * 4 + x] * W_rot[r * 4 + x];

    if (cc < CH) {                 // k: heads [nodes | aux | rot] = [0..7|8..15|16..23]
        float* kb = Kbuf + (size_t)i * KQROW + cc;
        kb[(0 + hh) * CH]  = nv;
        kb[(8 + hh) * CH]  = av;
        kb[(16 + hh) * CH] = rv;
    } else {
        const int c = cc - CH;
        QT[(size_t)((0 + hh) * CH + c) * S_LEN + i]  = nv;
        QT[(size_t)((8 + hh) * CH + c) * S_LEN + i]  = av;
        QT[(size_t)((16 + hh) * CH + c) * S_LEN + i] = rv;
    }
}

// ---------------------------------------------------------------------------
// Fused attention. Grid = 128 i-tiles. Block = 256 threads = 8 waves.
// Wave w processes heads {w, w+8, w+16} with online (flash) softmax,
// sums the 3 normalized head outputs in registers, then the 8 waves
// reduce deterministically through LDS.
// ---------------------------------------------------------------------------
__global__ __launch_bounds__(256) void attn_kernel(
        const float* __restrict__ Kbuf, const float* __restrict__ QT,
        const float* __restrict__ Vbuf, float* __restrict__ out) {
    __shared__ __align__(16) float pbuf[8][256];   // per-wave 16x16 P transpose tile
    __shared__ float obuf[16 * NDIM];              // cross-wave output reduction

    const int tid  = threadIdx.x;
    const int lane = tid & 31;
    const int wave = tid >> 5;
    const int hl   = lane >> 4;
    const int ln   = lane & 15;
    const int it   = blockIdx.x;
    float* pt = pbuf[wave];
    const float inv_sqrt_h = 0.35355339059327373f;   // 1/sqrt(8)

    v8f osum[8];
#pragma unroll
    for (int nc = 0; nc < 8; ++nc)
#pragma unroll
        for (int r = 0; r < 8; ++r) osum[nc][r] = 0.f;

    for (int hidx = 0; hidx < 3; ++hidx) {
        const int h = wave + 8 * hidx;

        // K tile as resident A operand (16 rows x 32 channels -> 8 K-slices)
        v2f ka[8];
        const float* Kp = Kbuf + (size_t)(it * 16 + ln) * KQROW + h * CH + 2 * hl;
#pragma unroll
        for (int s = 0; s < 8; ++s) ka[s] = *(const v2f*)(Kp + s * 4);

        v8f o[8];
        float m[8], l[8];
#pragma unroll
        for (int nc = 0; nc < 8; ++nc)
#pragma unroll
            for (int r = 0; r < 8; ++r) o[nc][r] = 0.f;
#pragma unroll
        for (int r = 0; r < 8; ++r) { m[r] = -__builtin_inff(); l[r] = 0.f; }

        const float* QTh = QT + (size_t)h * CH * S_LEN;

        for (int jt = 0; jt < S_LEN / 16; ++jt) {
            // ---- logits tile: D(16x16, i x j) = Ktile(16x32) @ Q^T(32x16)
            v8f p;
#pragma unroll
            for (int r = 0; r < 8; ++r) p[r] = 0.f;
#pragma unroll
            for (int s = 0; s < 8; ++s) {
                const float* qp = QTh + (size_t)(s * 4 + 2 * hl) * S_LEN + jt * 16 + ln;
                v2f qb;
                qb[0] = qp[0];
                qb[1] = qp[S_LEN];
                p = wmma4(ka[s], qb, p);
            }

            // ---- head transform + online softmax (row = r + 8*hl, across 16 lanes)
            float aval[8];
#pragma unroll
            for (int r = 0; r < 8; ++r) {
                float x = p[r];
                if (h >= 16) x = x * x;        // rot heads squared
                x *= inv_sqrt_h;
                float xm = x;
                xm = fmaxf(xm, __shfl_xor(xm, 1));
                xm = fmaxf(xm, __shfl_xor(xm, 2));
                xm = fmaxf(xm, __shfl_xor(xm, 4));
                xm = fmaxf(xm, __shfl_xor(xm, 8));   // row max, stays in half-wave
                const float mn    = fmaxf(m[r], xm);
                const float alpha = __expf(m[r] - mn);
                const float e     = __expf(x - mn);
                float es = e;
                es += __shfl_xor(es, 1);
                es += __shfl_xor(es, 2);
                es += __shfl_xor(es, 4);
                es += __shfl_xor(es, 8);             // row sum
                l[r] = l[r] * alpha + es;
                m[r] = mn;
                p[r] = e;
                aval[r] = alpha;
            }
            v8f av;
#pragma unroll
            for (int r = 0; r < 8; ++r) av[r] = aval[r];
#pragma unroll
            for (int nc = 0; nc < 8; ++nc) o[nc] *= av;

            // ---- P: C/D layout -> A layout via per-wave LDS tile (in-order DS)
#pragma unroll
            for (int r = 0; r < 8; ++r) pt[(r + 8 * hl) * 16 + ln] = p[r];
            __asm__ volatile("" ::: "memory");
            v2f pa[4];
            const float* pr = pt + ln * 16 + 2 * hl;
#pragma unroll
            for (int s = 0; s < 4; ++s) pa[s] = *(const v2f*)(pr + s * 4);
            __asm__ volatile("" ::: "memory");

            // ---- O(16x128) += P(16x16) @ V(16x128): 8 n-chunks x 4 K-slices
            const float* Vb = Vbuf + (size_t)(jt * 16) * VROW + h * NDIM + ln;
#pragma unroll
            for (int nc = 0; nc < 8; ++nc) {
#pragma unroll
                for (int s = 0; s < 4; ++s) {
                    const float* vp = Vb + (size_t)(s * 4 + 2 * hl) * VROW + nc * 16;
                    v2f vb;
                    vb[0] = vp[0];
                    vb[1] = vp[VROW];
                    o[nc] = wmma4(pa[s], vb, o[nc]);
                }
            }
        }

        // normalize by softmax denominator, accumulate into per-wave head sum
        v8f linv;
#pragma unroll
        for (int r = 0; r < 8; ++r) linv[r] = 1.0f / l[r];
#pragma unroll
        for (int nc = 0; nc < 8; ++nc) osum[nc] += o[nc] * linv;
    }

    // deterministic cross-wave reduction (ordered, no float atomics)
    for (int w = 0; w < 8; ++w) {
        if (wave == w) {
#pragma unroll
            for (int nc = 0; nc < 8; ++nc)
#pragma unroll
                for (int r = 0; r < 8; ++r) {
                    const int idx = (r + 8 * hl) * NDIM + nc * 16 + ln;
                    obuf[idx] = (w == 0) ? osum[nc][r] : obuf[idx] + osum[nc][r];
                }
        }
        __syncthreads();
    }
    for (int e = tid; e < 16 * NDIM; e += 256)
        out[(size_t)(it * 16 + (e >> 7)) * NDIM + (e & 127)] = obuf[e];
}

// ---------------------------------------------------------------------------
extern "C" void kernel_launch(void* const* d_in, const int* in_sizes, int n_in,
                              void* d_out, int out_size, void* d_ws, size_t ws_size,
                              hipStream_t stream) {
    const float* nodes   = (const float*)d_in[0];
    const float* aux     = (const float*)d_in[1];
    const float* rot     = (const float*)d_in[2];
    const float* W_nodes = (const float*)d_in[3];
    const float* b_nodes = (const float*)d_in[4];
    const float* W_aux   = (const float*)d_in[5];
    const float* b_aux   = (const float*)d_in[6];
    const float* W_rot   = (const float*)d_in[7];
    const float* W_val   = (const float*)d_in[8];
    const float* b_val   = (const float*)d_in[9];
    float* out = (float*)d_out;

    // workspace layout (floats): KQn(2048x512) | Kbuf(2048x768) | QT(768x2048) | Vbuf(2048x3072)
    float* ws   = (float*)d_ws;
    float* KQn  = ws;
    float* Kbuf = KQn + (size_t)S_LEN * 512;
    float* QT   = Kbuf + (size_t)S_LEN * KQROW;
    float* Vbuf = QT + (size_t)S_LEN * KQROW;

    // nodes -> kq (N=512): 32 n-tiles * 16 i-groups = 512 waves = 64 blocks
    gemm_wmma_kernel<<<64, 256, 0, stream>>>(nodes, W_nodes, b_nodes, KQn, 512);
    // nodes -> val (N=3072): 192 n-tiles * 16 i-groups = 3072 waves = 384 blocks
    gemm_wmma_kernel<<<384, 256, 0, stream>>>(nodes, W_val, b_val, Vbuf, VROW);
    // aux/rot projections + scatter into Kbuf / QT
    kq_small_kernel<<<(S_LEN * 512) / 256, 256, 0, stream>>>(
        aux, rot, W_aux, b_aux, W_rot, KQn, Kbuf, QT);
    // fused flash-style attention over all 24 heads
    attn_kernel<<<S_LEN / 16, 256, 0, stream>>>(Kbuf, QT, Vbuf, out);
}